// TransitionDown__28836410425490
// MI455X (gfx1250) — compile-verified
//
#include <hip/hip_runtime.h>
#include <hip/hip_bf16.h>

#define N_PTS 65536
#define M_CLUS 4096
#define KNN_K 16
#define C_IN 128
#define C_OUT 256
#define BN_EPS 1e-5f

typedef __attribute__((ext_vector_type(2))) float v2f;
typedef __attribute__((ext_vector_type(8))) float v8f;

// ---------------------------------------------------------------------------
// 1) Farthest point sampling: single workgroup, min_d lives in 256KB LDS.
// ---------------------------------------------------------------------------
__global__ __launch_bounds__(1024) void fps_kernel(const float* __restrict__ pos,
                                                   int* __restrict__ fps_idx) {
  extern __shared__ float min_d[];          // N_PTS floats (256 KB dynamic LDS)
  __shared__ float red_v[32];
  __shared__ int   red_i[32];
  __shared__ int   cur_s;

  const int t = threadIdx.x;
  for (int p = t; p < N_PTS; p += 1024) min_d[p] = 3.4e38f;
  if (t == 0) { fps_idx[0] = 0; cur_s = 0; }
  __syncthreads();

  for (int it = 1; it < M_CLUS; ++it) {
    const int cur = cur_s;
    const float qx = pos[cur * 3 + 0];
    const float qy = pos[cur * 3 + 1];
    const float qz = pos[cur * 3 + 2];

    float best = -1.0f;
    int   bidx = 0x7fffffff;
    for (int p = t; p < N_PTS; p += 1024) {
      float dx = pos[p * 3 + 0] - qx;
      float dy = pos[p * 3 + 1] - qy;
      float dz = pos[p * 3 + 2] - qz;
      float d  = dx * dx + dy * dy + dz * dz;
      float md = fminf(min_d[p], d);
      min_d[p] = md;
      if (md > best) { best = md; bidx = p; }
    }
    // wave32 argmax (tie -> lower index, matches jnp.argmax)
    #pragma unroll
    for (int off = 16; off > 0; off >>= 1) {
      float ov = __shfl_xor(best, off, 32);
      int   oi = __shfl_xor(bidx, off, 32);
      if (ov > best || (ov == best && oi < bidx)) { best = ov; bidx = oi; }
    }
    if ((t & 31) == 0) { red_v[t >> 5] = best; red_i[t >> 5] = bidx; }
    __syncthreads();
    if (t < 32) {
      best = red_v[t]; bidx = red_i[t];
      #pragma unroll
      for (int off = 16; off > 0; off >>= 1) {
        float ov = __shfl_xor(best, off, 32);
        int   oi = __shfl_xor(bidx, off, 32);
        if (ov > best || (ov == best && oi < bidx)) { best = ov; bidx = oi; }
      }
      if (t == 0) { cur_s = bidx; fps_idx[it] = bidx; }
    }
    __syncthreads();
  }
}

// ---------------------------------------------------------------------------
// 2) Gather sampled positions / batch ids.
// ---------------------------------------------------------------------------
__global__ void sub_gather_kernel(const float* __restrict__ pos,
                                  const int* __restrict__ batch,
                                  const int* __restrict__ fps_idx,
                                  float* __restrict__ sub_pos,
                                  int* __restrict__ sub_batch) {
  int m = blockIdx.x * blockDim.x + threadIdx.x;
  if (m >= M_CLUS) return;
  int idx = fps_idx[m];
  sub_pos[m * 3 + 0] = pos[idx * 3 + 0];
  sub_pos[m * 3 + 1] = pos[idx * 3 + 1];
  sub_pos[m * 3 + 2] = pos[idx * 3 + 2];
  sub_batch[m] = batch[idx];
}

// ---------------------------------------------------------------------------
// 3) GEMM h = x @ W + b with V_WMMA_F32_16X16X4_F32 (exact fp32 on the
//    matrix pipe). One wave -> one 16x16 output tile, 32 WMMAs over K=128.
//    A 16x4 layout: lanes 0-15 hold rows M=0..15 with K={k0,k0+1} in
//    {v0,v1}; lanes 16-31 hold K={k0+2,k0+3}.  B 4x16 mirrors this.
// ---------------------------------------------------------------------------
__global__ __launch_bounds__(256) void gemm_kernel(const float* __restrict__ x,
                                                   const float* __restrict__ W,
                                                   const float* __restrict__ bias,
                                                   float* __restrict__ h) {
  const int lane = threadIdx.x & 31;
  const int wave = blockIdx.x * 8 + (threadIdx.x >> 5);
  const int TN   = C_OUT / 16;              // 16 tiles along N
  const int tm   = wave / TN;
  const int tn   = wave % TN;
  const int row0 = tm * 16;
  const int col0 = tn * 16;
  const int half = lane >> 4;               // 0: K pair {0,1}, 1: K pair {2,3}
  const int lr   = lane & 15;

  const float* xa = x + (size_t)(row0 + lr) * C_IN + 2 * half;     // A stream
  const float* wb = W + (size_t)(2 * half) * C_OUT + col0 + lr;    // B stream

  v8f acc = {};
  #pragma unroll 8
  for (int k0 = 0; k0 < C_IN; k0 += 4) {
    v2f a = *(const v2f*)(xa + k0);         // x[row, k0+2h], x[row, k0+2h+1]
    v2f b;
    b.x = wb[(size_t)k0 * C_OUT];           // W[k0+2h,   col]
    b.y = wb[(size_t)(k0 + 1) * C_OUT];     // W[k0+2h+1, col]
    acc = __builtin_amdgcn_wmma_f32_16x16x4_f32(
        /*neg_a=*/false, a, /*neg_b=*/false, b,
        /*c_mod=*/(short)0, acc, /*reuse_a=*/false, /*reuse_b=*/false);
  }

  // C/D layout: VGPR i -> row (i + 8*half), column = lr
  const float bv = bias[col0 + lr];
  #pragma unroll
  for (int i = 0; i < 8; ++i) {
    int r = row0 + i + 8 * half;
    h[(size_t)r * C_OUT + col0 + lr] = acc[i] + bv;
  }
}

// ---------------------------------------------------------------------------
// 4) BatchNorm statistics: coalesced row-sweeps, per-column register
//    accumulators, one atomicAdd pair per (block, column).
// ---------------------------------------------------------------------------
__global__ void zero_acc_kernel(float* __restrict__ acc) {
  int t = threadIdx.x;
  if (t < 2 * C_OUT) acc[t] = 0.0f;
}

__global__ __launch_bounds__(256) void stats_kernel(const float* __restrict__ h,
                                                    float* __restrict__ acc) {
  const int c    = threadIdx.x;             // column
  const int row0 = blockIdx.x * 256;
  float s = 0.0f, ss = 0.0f;
  #pragma unroll 4
  for (int r = row0; r < row0 + 256; ++r) {
    float v = h[(size_t)r * C_OUT + c];
    s += v;
    ss += v * v;
  }
  atomicAdd(&acc[c], s);
  atomicAdd(&acc[C_OUT + c], ss);
}

__global__ void finalize_kernel(const float* __restrict__ acc,
                                const float* __restrict__ gamma,
                                const float* __restrict__ beta,
                                float* __restrict__ scale,
                                float* __restrict__ shift) {
  int c = threadIdx.x;
  float mu  = acc[c] * (1.0f / N_PTS);
  float var = acc[C_OUT + c] * (1.0f / N_PTS) - mu * mu;
  float s   = gamma[c] * rsqrtf(var + BN_EPS);
  scale[c] = s;
  shift[c] = beta[c] - mu * s;
}

// ---------------------------------------------------------------------------
// 5) KNN: one wave per query. Per-lane register-resident sorted top-16 over
//    2048 candidates, then 16 shuffle-min selection rounds to merge.
// ---------------------------------------------------------------------------
__global__ __launch_bounds__(256) void knn_kernel(const float* __restrict__ pos,
                                                  const float* __restrict__ sub_pos,
                                                  int* __restrict__ nbr) {
  const int lane = threadIdx.x & 31;
  const int wave = blockIdx.x * (blockDim.x >> 5) + (threadIdx.x >> 5);
  if (wave >= M_CLUS) return;

  const float qx = sub_pos[wave * 3 + 0];
  const float qy = sub_pos[wave * 3 + 1];
  const float qz = sub_pos[wave * 3 + 2];

  float bd[KNN_K];
  int   bi[KNN_K];
  #pragma unroll
  for (int j = 0; j < KNN_K; ++j) { bd[j] = 3.4e38f; bi[j] = 0x7fffffff; }

  for (int p = lane; p < N_PTS; p += 32) {
    float dx = pos[p * 3 + 0] - qx;
    float dy = pos[p * 3 + 1] - qy;
    float dz = pos[p * 3 + 2] - qz;
    float d  = dx * dx + dy * dy + dz * dz;
    if (d < bd[KNN_K - 1]) {
      int j = KNN_K - 1;
      #pragma unroll
      for (int u = 0; u < KNN_K - 1; ++u) {
        if (j > 0 && bd[j - 1] > d) { bd[j] = bd[j - 1]; bi[j] = bi[j - 1]; --j; }
      }
      bd[j] = d; bi[j] = p;
    }
  }

  // 16 rounds of wave-wide min selection (tie -> lower index == top_k order)
  int head = 0;
  for (int r = 0; r < KNN_K; ++r) {
    float v = (head < KNN_K) ? bd[head] : 3.4e38f;
    int   i = (head < KNN_K) ? bi[head] : 0x7fffffff;
    float mv = v; int mi = i;
    #pragma unroll
    for (int off = 16; off > 0; off >>= 1) {
      float ov = __shfl_xor(mv, off, 32);
      int   oi = __shfl_xor(mi, off, 32);
      if (ov < mv || (ov == mv && oi < mi)) { mv = ov; mi = oi; }
    }
    if (head < KNN_K && mi == i && mv == v) head++;   // unique owner advances
    if (lane == 0) nbr[wave * KNN_K + r] = mi;
  }
}

// ---------------------------------------------------------------------------
// 6) Gather + max-pool with BN+ReLU folded in. relu(s*x+t) is monotone in x
//    with sign(s): take max (s>=0) or min (s<0) of raw h, apply affine once.
// ---------------------------------------------------------------------------
__global__ __launch_bounds__(256) void gather_max_kernel(const float* __restrict__ h,
                                                         const int* __restrict__ nbr,
                                                         const float* __restrict__ scale,
                                                         const float* __restrict__ shift,
                                                         float* __restrict__ out) {
  const int m = blockIdx.x;
  const int c = threadIdx.x;
  const float s = scale[c];
  const float t = shift[c];
  const bool pos_s = (s >= 0.0f);
  float best = pos_s ? -3.4e38f : 3.4e38f;
  #pragma unroll
  for (int k = 0; k < KNN_K; ++k) {
    int n = nbr[m * KNN_K + k];
    float v = h[(size_t)n * C_OUT + c];
    best = pos_s ? fmaxf(best, v) : fminf(best, v);
  }
  out[(size_t)m * C_OUT + c] = fmaxf(s * best + t, 0.0f);
}

// ---------------------------------------------------------------------------
// Launch wiring
// ---------------------------------------------------------------------------
extern "C" void kernel_launch(void* const* d_in, const int* in_sizes, int n_in,
                              void* d_out, int out_size, void* d_ws, size_t ws_size,
                              hipStream_t stream) {
  const float* x     = (const float*)d_in[0];
  const float* pos   = (const float*)d_in[1];
  const int*   batch = (const int*)d_in[2];
  const float* W     = (const float*)d_in[3];
  const float* b     = (const float*)d_in[4];
  const float* gamma = (const float*)d_in[5];
  const float* beta  = (const float*)d_in[6];

  // workspace layout
  float* h     = (float*)d_ws;                      // N_PTS * C_OUT
  float* acc   = h + (size_t)N_PTS * C_OUT;         // 2 * C_OUT
  float* scale = acc + 2 * C_OUT;                   // C_OUT
  float* shift = scale + C_OUT;                     // C_OUT
  int* fps_idx = (int*)(shift + C_OUT);             // M_CLUS
  int* nbr     = fps_idx + M_CLUS;                  // M_CLUS * KNN_K

  // output layout: out [M, C_OUT] ++ sub_pos [M, 3] ++ sub_batch [M]
  float* out       = (float*)d_out;
  float* sub_pos   = out + (size_t)M_CLUS * C_OUT;
  int*   sub_batch = (int*)(sub_pos + (size_t)M_CLUS * 3);

  // 1) FPS (serial bottleneck; min_d in 256KB LDS)
  fps_kernel<<<1, 1024, N_PTS * sizeof(float), stream>>>(pos, fps_idx);

  // 2) sampled pos / batch
  sub_gather_kernel<<<(M_CLUS + 255) / 256, 256, 0, stream>>>(pos, batch, fps_idx,
                                                              sub_pos, sub_batch);

  // 3) WMMA GEMM: (N/16)*(C_OUT/16) = 65536 waves, 8 waves/block
  gemm_kernel<<<(N_PTS / 16) * (C_OUT / 16) / 8, 256, 0, stream>>>(x, W, b, h);

  // 4) BN statistics
  zero_acc_kernel<<<1, 512, 0, stream>>>(acc);
  stats_kernel<<<N_PTS / 256, 256, 0, stream>>>(h, acc);
  finalize_kernel<<<1, C_OUT, 0, stream>>>(acc, gamma, beta, scale, shift);

  // 5) KNN: one wave per query
  knn_kernel<<<M_CLUS / 8, 256, 0, stream>>>(pos, sub_pos, nbr);

  // 6) gather + fused BN/ReLU max-pool
  gather_max_kernel<<<M_CLUS, 256, 0, stream>>>(h, nbr, scale, shift, out);
}